// GraphConv2d_33947421508468
// MI455X (gfx1250) — compile-verified
//
#include <hip/hip_runtime.h>

typedef __attribute__((ext_vector_type(2))) float v2f;
typedef __attribute__((ext_vector_type(8))) float v8f;

#define B_    16
#define C_    192
#define N_    3136
#define K_    9
#define OUT_  384
#define CIN_G 96           // 2*C/GROUPS
#define MT_   (OUT_ / 16)  // 24 m-tiles
#define NT_   (N_ / 16)    // 196 n-tiles
#define LDH   17           // padded LDS row stride (floats)

__global__ __launch_bounds__(256)
void graphconv_fused_wmma(const float* __restrict__ x,
                          const int*   __restrict__ edge,
                          const float* __restrict__ w,
                          const float* __restrict__ bias,
                          float*       __restrict__ out)
{
    __shared__ float sh[2 * C_ * LDH];   // interleaved h tile: 384 rows x 16 cols (padded)
    __shared__ int   sii[16 * K_];       // x_i indices (edge_index[1])
    __shared__ int   sij[16 * K_];       // x_j indices (edge_index[0])

    const int tid = threadIdx.x;
    const int blk = blockIdx.x;
    const int b   = blk / NT_;
    const int nb  = (blk % NT_) * 16;

    // ---- stage neighbor indices for this 16-column tile ----
    if (tid < 16 * K_) {
        const int nl = tid / K_;
        const int k  = tid % K_;
        const int n  = nb + nl;
        sij[tid] = edge[((size_t)(0 * B_ + b) * N_ + n) * K_ + k];
        sii[tid] = edge[((size_t)(1 * B_ + b) * N_ + n) * K_ + k];
    }
    __syncthreads();

    // ---- gather + max-relative + channel-interleave into LDS ----
    {
        const int nl = tid & 15;        // column within tile
        const int cl = tid >> 4;        // 0..15 channel sub-index
        const float* featb = x + (size_t)b * C_ * N_;
        for (int ci = 0; ci < C_ / 16; ++ci) {
            const int c = ci * 16 + cl;
            const float* fc = featb + (size_t)c * N_;
            const float xc = fc[nb + nl];
            float m = -3.402823466e38f;
            #pragma unroll
            for (int k = 0; k < K_; ++k) {
                const float xj = fc[sij[nl * K_ + k]];
                const float xi = fc[sii[nl * K_ + k]];
                const float d  = xj - xi;
                m = (d > m) ? d : m;
            }
            sh[(2 * c)     * LDH + nl] = xc;   // even h-channel: x
            sh[(2 * c + 1) * LDH + nl] = m;    // odd  h-channel: max_k(x_j - x_i)
        }
    }
    __syncthreads();

    // ---- grouped GEMM via V_WMMA_F32_16X16X4_F32 ----
    const int wave = tid >> 5;       // 0..7
    const int lane = tid & 31;
    const int half = lane >> 4;      // 0: K={0,1}, 1: K={2,3}
    const int l16  = lane & 15;
    const int kofs = half * 2;

    for (int mt = wave; mt < MT_; mt += 8) {
        const int g     = mt / 6;          // 6 m-tiles per group (96/16)
        const int obase = mt * 16;
        v8f acc = {0.f, 0.f, 0.f, 0.f, 0.f, 0.f, 0.f, 0.f};

        const float* wr = w + (size_t)(obase + l16) * CIN_G;  // A: this lane's W row
        const float* hb = &sh[(g * CIN_G) * LDH + l16];       // B: group's h rows, this column

        #pragma unroll 4
        for (int k0 = 0; k0 < CIN_G; k0 += 4) {
            v2f a, bf;
            a.x  = wr[k0 + kofs];               // contiguous pair -> b64 load
            a.y  = wr[k0 + kofs + 1];
            bf.x = hb[(size_t)(k0 + kofs)     * LDH];
            bf.y = hb[(size_t)(k0 + kofs + 1) * LDH];
            acc = __builtin_amdgcn_wmma_f32_16x16x4_f32(
                      false, a, false, bf, (short)0, acc, false, false);
        }

        // epilogue: bias + ReLU + store (D layout: VGPR r -> row r / r+8)
        #pragma unroll
        for (int r = 0; r < 8; ++r) {
            const int o = obase + r + half * 8;
            const int n = nb + l16;
            float v = acc[r] + bias[o];
            v = (v > 0.f) ? v : 0.f;
            out[((size_t)b * OUT_ + o) * N_ + n] = v;
        }
    }
}

extern "C" void kernel_launch(void* const* d_in, const int* in_sizes, int n_in,
                              void* d_out, int out_size, void* d_ws, size_t ws_size,
                              hipStream_t stream) {
    const float* x    = (const float*)d_in[0];   // (16, 192, 3136, 1) f32
    const int*   edge = (const int*)  d_in[1];   // (2, 16, 3136, 9) i32
    const float* w    = (const float*)d_in[2];   // (384, 96) f32
    const float* bias = (const float*)d_in[3];   // (384,) f32
    float*       out  = (float*)d_out;           // (16, 384, 3136, 1) f32

    const int grid = B_ * NT_;                   // 3136 blocks
    graphconv_fused_wmma<<<grid, 256, 0, stream>>>(x, edge, w, bias, out);
}